// BFP_37426345017360
// MI455X (gfx1250) — compile-verified
//
#include <hip/hip_runtime.h>
#include <hip/hip_bf16.h>

typedef __attribute__((ext_vector_type(2))) float v2f;
typedef __attribute__((ext_vector_type(8))) float v8f;

#define C_CH   256
#define HW     4096      // 64*64
#define NEG_INF (-3.402823466e38f)

#define WMMA_F32(a, b, c) \
    __builtin_amdgcn_wmma_f32_16x16x4_f32(false, (a), false, (b), (short)0, (c), false, false)

// ---------------------------------------------------------------------------
// Phase A: gather -> bsf[n,c,64,64] = ( maxpool2(x0) + x1 + up2(x2) + up4(x3) ) / 4
// ---------------------------------------------------------------------------
__global__ void gather_kernel(const float* __restrict__ x0, const float* __restrict__ x1,
                              const float* __restrict__ x2, const float* __restrict__ x3,
                              float* __restrict__ bsf) {
    int idx = blockIdx.x * blockDim.x + threadIdx.x;          // 2*256*64*64 = 2097152
    int ox = idx & 63, oy = (idx >> 6) & 63, c = (idx >> 12) & 255, n = idx >> 20;
    const float* p0 = x0 + (((size_t)(n * 256 + c) * 128 + 2 * oy) * 128 + 2 * ox);
    float m = fmaxf(fmaxf(p0[0], p0[1]), fmaxf(p0[128], p0[129]));
    float v1 = x1[((n * 256 + c) * 64 + oy) * 64 + ox];
    float v2 = x2[((n * 256 + c) * 32 + (oy >> 1)) * 32 + (ox >> 1)];
    float v3 = x3[((n * 256 + c) * 16 + (oy >> 2)) * 16 + (ox >> 2)];
    bsf[idx] = (m + v1 + v2 + v3) * 0.25f;
}

// ---------------------------------------------------------------------------
// WMMA fp32 GEMM core: 2x2 register-blocked, 32x32 output tile per wave.
// v_wmma_f32_16x16x4_f32 fragment layouts (ISA 7.12.2):
//   A 16x4: lane<16 -> (M=lane, K=kk+{0,1}), lane>=16 -> (M=lane-16, K=kk+{2,3})
//   B 4x16: lane<16 -> (K=kk+{0,1}, N=lane), lane>=16 -> (K=kk+{2,3}, N=lane-16)
//   C/D:    vgpr r, lane<16 -> (M=r, N=lane); lane>=16 -> (M=8+r, N=lane-16)
// AT: A[m,k] = A[k*lda + m] (transposed source)   else A[m*lda + k]
// BT: B[k,n] = B[n*ldb + k] (transposed source)   else B[k*ldb + n]
// ---------------------------------------------------------------------------
template <bool AT, bool BT>
__device__ __forceinline__ void wmma_core_2x2(const float* __restrict__ A, int lda,
                                              const float* __restrict__ B, int ldb,
                                              int K, int m0, int n0, int l, int h,
                                              v8f acc[2][2]) {
    for (int kk = 0; kk < K; kk += 4) {
        int kb = kk + h * 2;
        v2f a0, a1, b0, b1;
        if (AT) {
            a0.x = A[kb * lda + m0 + l];       a0.y = A[(kb + 1) * lda + m0 + l];
            a1.x = A[kb * lda + m0 + 16 + l];  a1.y = A[(kb + 1) * lda + m0 + 16 + l];
        } else {
            a0.x = A[(m0 + l) * lda + kb];       a0.y = A[(m0 + l) * lda + kb + 1];
            a1.x = A[(m0 + 16 + l) * lda + kb];  a1.y = A[(m0 + 16 + l) * lda + kb + 1];
        }
        if (BT) {
            b0.x = B[(n0 + l) * ldb + kb];       b0.y = B[(n0 + l) * ldb + kb + 1];
            b1.x = B[(n0 + 16 + l) * ldb + kb];  b1.y = B[(n0 + 16 + l) * ldb + kb + 1];
        } else {
            b0.x = B[kb * ldb + n0 + l];       b0.y = B[(kb + 1) * ldb + n0 + l];
            b1.x = B[kb * ldb + n0 + 16 + l];  b1.y = B[(kb + 1) * ldb + n0 + 16 + l];
        }
        acc[0][0] = WMMA_F32(a0, b0, acc[0][0]);
        acc[0][1] = WMMA_F32(a0, b1, acc[0][1]);
        acc[1][0] = WMMA_F32(a1, b0, acc[1][0]);
        acc[1][1] = WMMA_F32(a1, b1, acc[1][1]);
    }
}

// G/T/P = W[256x256] * X[256x4096] + bias[row]
__global__ void gemm_wx(const float* __restrict__ W, const float* __restrict__ X,
                        const float* __restrict__ bias, float* __restrict__ D) {
    int wave = blockIdx.x * (blockDim.x >> 5) + (threadIdx.x >> 5);
    int lane = threadIdx.x & 31, l = lane & 15, h = lane >> 4;
    int mt = wave >> 7, nt = wave & 127;           // 8 x 128 tiles of 32x32
    int m0 = mt * 32, n0 = nt * 32;
    v8f acc[2][2] = {};
    wmma_core_2x2<false, false>(W, C_CH, X, HW, C_CH, m0, n0, l, h, acc);
    for (int mi = 0; mi < 2; ++mi)
        for (int ni = 0; ni < 2; ++ni) {
            int rb = m0 + mi * 16 + h * 8, col = n0 + ni * 16 + l;
            for (int r = 0; r < 8; ++r)
                D[(rb + r) * HW + col] = acc[mi][ni][r] + bias[rb + r];
        }
}

// S[4096x4096] = T^T * P  (T,P are [256 x 4096] channel-major), K = 256
__global__ void gemm_attn(const float* __restrict__ T, const float* __restrict__ P,
                          float* __restrict__ S) {
    int wave = blockIdx.x * (blockDim.x >> 5) + (threadIdx.x >> 5);
    int lane = threadIdx.x & 31, l = lane & 15, h = lane >> 4;
    int mt = wave >> 7, nt = wave & 127;           // 128 x 128 tiles of 32x32
    int m0 = mt * 32, n0 = nt * 32;
    v8f acc[2][2] = {};
    wmma_core_2x2<true, false>(T, HW, P, HW, C_CH, m0, n0, l, h, acc);
    for (int mi = 0; mi < 2; ++mi)
        for (int ni = 0; ni < 2; ++ni) {
            int rb = m0 + mi * 16 + h * 8, col = n0 + ni * 16 + l;
            for (int r = 0; r < 8; ++r)
                S[(size_t)(rb + r) * HW + col] = acc[mi][ni][r];
        }
}

// row-wise softmax over 4096 columns, in place; one block per row
__global__ void softmax_rows(float* __restrict__ S) {
    __shared__ float red[256];
    float* p = S + (size_t)blockIdx.x * HW;
    int tid = threadIdx.x;
    float mx = NEG_INF;
    for (int j = tid; j < HW; j += 256) mx = fmaxf(mx, p[j]);
    red[tid] = mx; __syncthreads();
    for (int s = 128; s > 0; s >>= 1) { if (tid < s) red[tid] = fmaxf(red[tid], red[tid + s]); __syncthreads(); }
    mx = red[0]; __syncthreads();
    float sum = 0.f;
    for (int j = tid; j < HW; j += 256) { float e = __expf(p[j] - mx); p[j] = e; sum += e; }
    red[tid] = sum; __syncthreads();
    for (int s = 128; s > 0; s >>= 1) { if (tid < s) red[tid] += red[tid + s]; __syncthreads(); }
    float inv = 1.0f / red[0];
    for (int j = tid; j < HW; j += 256) p[j] *= inv;
}

// Y[256x4096] = G * S^T : Y[c,q] = sum_k G[c,k] * S[q,k], K = 4096
__global__ void gemm_y(const float* __restrict__ G, const float* __restrict__ S,
                       float* __restrict__ Y) {
    int wave = blockIdx.x * (blockDim.x >> 5) + (threadIdx.x >> 5);
    int lane = threadIdx.x & 31, l = lane & 15, h = lane >> 4;
    int mt = wave >> 7, nt = wave & 127;           // 8 x 128 tiles of 32x32
    int m0 = mt * 32, n0 = nt * 32;
    v8f acc[2][2] = {};
    wmma_core_2x2<false, true>(G, HW, S, HW, HW, m0, n0, l, h, acc);
    for (int mi = 0; mi < 2; ++mi)
        for (int ni = 0; ni < 2; ++ni) {
            int rb = m0 + mi * 16 + h * 8, col = n0 + ni * 16 + l;
            for (int r = 0; r < 8; ++r)
                Y[(rb + r) * HW + col] = acc[mi][ni][r];
        }
}

// BO[256x4096] = conv_out_w * Y + bias[row] + bsf   (fused non-local residual)
__global__ void gemm_out(const float* __restrict__ W, const float* __restrict__ Y,
                         const float* __restrict__ bias, const float* __restrict__ bsf,
                         float* __restrict__ BO) {
    int wave = blockIdx.x * (blockDim.x >> 5) + (threadIdx.x >> 5);
    int lane = threadIdx.x & 31, l = lane & 15, h = lane >> 4;
    int mt = wave >> 7, nt = wave & 127;
    int m0 = mt * 32, n0 = nt * 32;
    v8f acc[2][2] = {};
    wmma_core_2x2<false, false>(W, C_CH, Y, HW, C_CH, m0, n0, l, h, acc);
    for (int mi = 0; mi < 2; ++mi)
        for (int ni = 0; ni < 2; ++ni) {
            int rb = m0 + mi * 16 + h * 8, col = n0 + ni * 16 + l;
            for (int r = 0; r < 8; ++r) {
                int off = (rb + r) * HW + col;
                BO[off] = acc[mi][ni][r] + bias[rb + r] + bsf[off];
            }
        }
}

// ---------------------------------------------------------------------------
// Phase C: scatter back with residual adds
// ---------------------------------------------------------------------------
__global__ void scatter0(const float* __restrict__ BO, const float* __restrict__ x0,
                         float* __restrict__ out) {          // 128x128: nearest up2
    int idx = blockIdx.x * blockDim.x + threadIdx.x;         // 8388608
    int x = idx & 127, y = (idx >> 7) & 127, c = (idx >> 14) & 255, n = idx >> 22;
    float r = BO[(size_t)(n * 256 + c) * HW + (y >> 1) * 64 + (x >> 1)];
    out[idx] = r + x0[idx];
}
__global__ void scatter1(const float* __restrict__ BO, const float* __restrict__ x1,
                         float* __restrict__ out) {          // 64x64: identity
    int idx = blockIdx.x * blockDim.x + threadIdx.x;         // 2097152
    out[idx] = BO[idx] + x1[idx];
}
__global__ void scatter2(const float* __restrict__ BO, const float* __restrict__ x2,
                         float* __restrict__ out) {          // 32x32: maxpool2
    int idx = blockIdx.x * blockDim.x + threadIdx.x;         // 524288
    int x = idx & 31, y = (idx >> 5) & 31, c = (idx >> 10) & 255, n = idx >> 18;
    const float* p = BO + (size_t)(n * 256 + c) * HW + (2 * y) * 64 + 2 * x;
    float m = fmaxf(fmaxf(p[0], p[1]), fmaxf(p[64], p[65]));
    out[idx] = m + x2[idx];
}
__global__ void scatter3(const float* __restrict__ BO, const float* __restrict__ x3,
                         float* __restrict__ out) {          // 16x16: maxpool4
    int idx = blockIdx.x * blockDim.x + threadIdx.x;         // 131072
    int x = idx & 15, y = (idx >> 4) & 15, c = (idx >> 8) & 255, n = idx >> 16;
    const float* p = BO + (size_t)(n * 256 + c) * HW + (4 * y) * 64 + 4 * x;
    float m = NEG_INF;
    for (int dy = 0; dy < 4; ++dy)
        for (int dx = 0; dx < 4; ++dx)
            m = fmaxf(m, p[dy * 64 + dx]);
    out[idx] = m + x3[idx];
}

// ---------------------------------------------------------------------------
extern "C" void kernel_launch(void* const* d_in, const int* in_sizes, int n_in,
                              void* d_out, int out_size, void* d_ws, size_t ws_size,
                              hipStream_t stream) {
    const float* x0      = (const float*)d_in[0];
    const float* x1      = (const float*)d_in[1];
    const float* x2      = (const float*)d_in[2];
    const float* x3      = (const float*)d_in[3];
    const float* g_w     = (const float*)d_in[4];
    const float* g_b     = (const float*)d_in[5];
    const float* theta_w = (const float*)d_in[6];
    const float* theta_b = (const float*)d_in[7];
    const float* phi_w   = (const float*)d_in[8];
    const float* phi_b   = (const float*)d_in[9];
    const float* co_w    = (const float*)d_in[10];
    const float* co_b    = (const float*)d_in[11];
    float* out = (float*)d_out;

    // workspace layout (floats); total ~25.2M floats = ~101 MB
    float* ws  = (float*)d_ws;
    const size_t PB = (size_t)C_CH * HW;      // 1048576 floats per batch plane
    float* bsf = ws;                          // 2*PB
    float* G   = ws + 2 * PB;                 // PB (per batch, reused)
    float* T   = ws + 3 * PB;
    float* P   = ws + 4 * PB;
    float* Y   = ws + 5 * PB;
    float* BO  = ws + 6 * PB;                 // 2*PB
    float* S   = ws + 8 * PB;                 // 16777216 floats (64 MB, reused per batch, L2-resident)

    gather_kernel<<<8192, 256, 0, stream>>>(x0, x1, x2, x3, bsf);

    for (int n = 0; n < 2; ++n) {
        const float* X = bsf + (size_t)n * PB;
        gemm_wx  <<<128,  256, 0, stream>>>(g_w,     X, g_b,     G);    // 1024 tile-waves
        gemm_wx  <<<128,  256, 0, stream>>>(theta_w, X, theta_b, T);
        gemm_wx  <<<128,  256, 0, stream>>>(phi_w,   X, phi_b,   P);
        gemm_attn<<<2048, 256, 0, stream>>>(T, P, S);                   // 16384 tile-waves
        softmax_rows<<<4096, 256, 0, stream>>>(S);
        gemm_y   <<<128,  256, 0, stream>>>(G, S, Y);
        gemm_out <<<128,  256, 0, stream>>>(co_w, Y, co_b, X, BO + (size_t)n * PB);
    }

    scatter0<<<32768, 256, 0, stream>>>(BO, x0, out);                    // 8388608 elems
    scatter1<<< 8192, 256, 0, stream>>>(BO, x1, out + 8388608);          // 2097152
    scatter2<<< 2048, 256, 0, stream>>>(BO, x2, out + 10485760);         //  524288
    scatter3<<<  512, 256, 0, stream>>>(BO, x3, out + 11010048);         //  131072
}